// AttBlock_45870250721474
// MI455X (gfx1250) — compile-verified
//
#include <hip/hip_runtime.h>

typedef __attribute__((ext_vector_type(4))) float v4f;

#define N_      64
#define C_      256
#define H_      56
#define W_      56
#define HW      (H_ * W_)        // 3136
#define HW4     (HW / 4)         // 784 float4 per plane
#define HS_     28
#define WS_     28
#define NDEMOG_ 4
#define CPB     8                // channels per block
#define NCB     (C_ / CPB)       // 32 channel-groups
#define Y_SIZE  (N_ * C_ * HW)   // 51,380,224
#define AC_SIZE (NDEMOG_ * C_)   // 1,024
#define ASP_SIZE (NDEMOG_ * HW)  // 12,544

__device__ __forceinline__ float sigmoidf(float v) {
    return 1.0f / (1.0f + __expf(-v));
}

// Materialize the two small gate outputs into the tail of d_out:
//   ac[d,c]    = sigmoid(att_channel[d,c])                      (1,024 elems)
//   asp[d,h,w] = sigmoid(att_spatial[d, h*28/56, w*28/56])      (12,544 elems, nearest-upsample 2x)
__global__ void att_setup_kernel(const float* __restrict__ att_channel,
                                 const float* __restrict__ att_spatial,
                                 float* __restrict__ ac_out,
                                 float* __restrict__ asp_out) {
    int idx = blockIdx.x * blockDim.x + threadIdx.x;
    if (idx < AC_SIZE) {
        ac_out[idx] = sigmoidf(att_channel[idx]);
    } else if (idx < AC_SIZE + ASP_SIZE) {
        int j = idx - AC_SIZE;          // (d, h, w) flattened over 4*56*56
        int d  = j / HW;
        int hw = j % HW;
        int h  = hw / W_;
        int w  = hw % W_;
        int ih = (h * HS_) / H_;        // == h/2
        int iw = (w * WS_) / W_;        // == w/2
        asp_out[j] = sigmoidf(att_spatial[(d * HS_ + ih) * WS_ + iw]);
    }
}

// Main gating kernel: one block per (n, 8-channel group).
// Stages asp[d] (12.5 KB) into LDS via gfx1250 async global->LDS b128 copies,
// then streams x -> y with non-temporal 128-bit accesses (keeps gate tables in L2).
__global__ void __launch_bounds__(256)
att_apply_kernel(const float* __restrict__ x,
                 const int* __restrict__ demog,
                 const float* __restrict__ ac,
                 const float* __restrict__ asp,
                 float* __restrict__ y) {
    __shared__ v4f s_asp[HW4];         // 12,544 bytes

    const int bid = blockIdx.x;
    const int n   = bid / NCB;
    const int cg  = bid % NCB;
    const int tid = threadIdx.x;
    const int d   = demog[n];          // uniform per block -> scalar load

    // ---- async copy asp[d] (784 x b128) global -> LDS (ASYNCcnt-tracked) ----
    const v4f* asp4 = (const v4f*)(asp + (long)d * HW);
    for (int i = tid; i < HW4; i += 256) {
        // Low 32 bits of the flat shared-aperture address are the LDS byte offset.
        unsigned lds_addr = (unsigned)(unsigned long long)(&s_asp[i]);
        const v4f* gptr = asp4 + i;
        asm volatile("global_load_async_to_lds_b128 %0, %1, off"
                     :
                     : "v"(lds_addr), "v"(gptr)
                     : "memory");
    }
    asm volatile("s_wait_asynccnt 0" ::: "memory");
    __syncthreads();

    // ---- stream 8 channel planes: y = x * gc * asp ----
    const int  c0     = cg * CPB;
    const long plane0 = ((long)n * C_ + c0) * HW;
    const v4f* xb = (const v4f*)(x + plane0);
    v4f*       yb = (v4f*)(y + plane0);

    #pragma unroll
    for (int c = 0; c < CPB; ++c) {
        const float gc = ac[d * C_ + c0 + c];
        const v4f* xp = xb + c * HW4;
        v4f*       yp = yb + c * HW4;
        for (int i = tid; i < HW4; i += 256) {
            v4f xv = __builtin_nontemporal_load(xp + i);   // TH_NT: no reuse of x
            v4f sv = s_asp[i];                             // ds_load_b128, reused 8x
            v4f r  = xv * (sv * gc);
            __builtin_nontemporal_store(r, yp + i);        // TH_NT: y never re-read
        }
    }
}

extern "C" void kernel_launch(void* const* d_in, const int* in_sizes, int n_in,
                              void* d_out, int out_size, void* d_ws, size_t ws_size,
                              hipStream_t stream) {
    const float* x           = (const float*)d_in[0];
    const int*   demog       = (const int*)d_in[1];
    const float* att_channel = (const float*)d_in[2];
    const float* att_spatial = (const float*)d_in[3];

    float* y       = (float*)d_out;
    float* ac_out  = y + Y_SIZE;          // outputs 2 and 3 live in the d_out tail
    float* asp_out = ac_out + AC_SIZE;

    // 1) build gate tables (also required outputs)
    const int setup_elems = AC_SIZE + ASP_SIZE;
    att_setup_kernel<<<(setup_elems + 255) / 256, 256, 0, stream>>>(
        att_channel, att_spatial, ac_out, asp_out);

    // 2) bandwidth-bound gating stream: 2048 blocks x 256 threads (8 wave32)
    att_apply_kernel<<<N_ * NCB, 256, 0, stream>>>(x, demog, ac_out, asp_out, y);
}